// Sampler_69922067578951
// MI455X (gfx1250) — compile-verified
//
#include <hip/hip_runtime.h>
#include <hip/hip_bf16.h>
#include <stdint.h>

// Categorical sampling via Gumbel-max: out[b] = argmax_v logits[b][v]/T[b] + G
// B=256 rows, V=128000 cols, f32. Bandwidth-bound: 131 MB @ 23.3 TB/s ~ 5.6us.
// Streaming path: 3-stage ring of GLOBAL_LOAD_ASYNC_TO_LDS_B128 (ASYNCcnt)
// with non-temporal hint, so HBM fetch of tiles k+1/k+2 overlaps VALU
// consumption of tile k from LDS.

#define BLOCK  256                  // 8 wave32 per workgroup
#define CHUNKS 5                    // b128 async loads per thread per tile
#define TILE   (BLOCK * 4 * CHUNKS) // 5120 floats = 20 KB per tile
#define PIPE   3                    // pipeline depth (60 KB LDS ring)

__device__ __forceinline__ unsigned hash32(unsigned x) {
    x ^= x >> 16; x *= 0x7feb352dU;
    x ^= x >> 15; x *= 0x846ca68bU;
    x ^= x >> 16;
    return x;
}

// Gumbel(0,1) noise from 32-bit hash: u in (0,1), g = -log(-log(u)).
__device__ __forceinline__ float gumbel(unsigned h) {
    float u = (float)(h >> 8) * 0x1.0p-24f + 0x1.0p-25f;
    return -__logf(-__logf(u));
}

// Issue CHUNKS async b128 copies global->LDS for this thread's slice of a
// tile. ASYNCcnt increments once per wave per instruction -> +CHUNKS per tile.
// Data is read exactly once -> non-temporal hint keeps WGP$/L2 clean.
__device__ __forceinline__ void async_copy_tile(const float* __restrict__ gsrc,
                                                float* ldst, int t) {
#pragma unroll
    for (int j = 0; j < CHUNKS; ++j) {
        const float* gp = gsrc + j * (BLOCK * 4) + 4 * t;
        // Flat pointer to __shared__ has the LDS byte offset in its low 32
        // bits (AS3 -> flat addrspacecast = {shared_aperture_hi, offset32}).
        unsigned lp = (unsigned)(uintptr_t)(ldst + j * (BLOCK * 4) + 4 * t);
        asm volatile("global_load_async_to_lds_b128 %0, %1, off th:TH_LOAD_NT"
                     :: "v"(lp), "v"(gp)
                     : "memory");
    }
}

__global__ __launch_bounds__(BLOCK)
void Sampler_69922067578951_kernel(const float* __restrict__ logits,
                                   const float* __restrict__ temps,
                                   int* __restrict__ out, int V) {
    __shared__ float tile[PIPE][TILE];  // 60 KB staging ring
    __shared__ float redv[BLOCK];
    __shared__ int   redi[BLOCK];

    const int b = blockIdx.x;
    const int t = threadIdx.x;
    const float invT = 1.0f / temps[b];
    const float* __restrict__ row = logits + (size_t)b * (size_t)V;
    const unsigned bV = (unsigned)b * (unsigned)V;

    float best = -INFINITY;
    int bestIdx = 0;

    const int ntiles = V / TILE;
    if (ntiles > 0) {
        // Prologue: fill up to PIPE-1 stages ahead.
        async_copy_tile(row, &tile[0][0], t);
        if (ntiles > 1)
            async_copy_tile(row + (size_t)TILE, &tile[1][0], t);

        for (int k = 0; k < ntiles; ++k) {
            const int buf = k % PIPE;
            // Keep PIPE tiles in flight; wait only until tile k has landed
            // (per-wave async loads complete in order).
            if (k + 2 < ntiles) {
                async_copy_tile(row + (size_t)(k + 2) * TILE,
                                &tile[(k + 2) % PIPE][0], t);
                asm volatile("s_wait_asynccnt 10" ::: "memory");
            } else if (k + 1 < ntiles) {
                asm volatile("s_wait_asynccnt 5" ::: "memory");
            } else {
                asm volatile("s_wait_asynccnt 0" ::: "memory");
            }
            __syncthreads();   // all 8 waves' slices of tile k are in LDS

            const int base = k * TILE;
#pragma unroll
            for (int j = 0; j < CHUNKS; ++j) {
                const int off = j * (BLOCK * 4) + 4 * t;
                const float4 x = *(const float4*)&tile[buf][off];  // ds_load_b128
                const int v0 = base + off;
                const unsigned id = bV + (unsigned)v0;
                float s;
                s = fmaf(x.x, invT, gumbel(hash32(id + 0u)));
                if (s > best) { best = s; bestIdx = v0 + 0; }
                s = fmaf(x.y, invT, gumbel(hash32(id + 1u)));
                if (s > best) { best = s; bestIdx = v0 + 1; }
                s = fmaf(x.z, invT, gumbel(hash32(id + 2u)));
                if (s > best) { best = s; bestIdx = v0 + 2; }
                s = fmaf(x.w, invT, gumbel(hash32(id + 3u)));
                if (s > best) { best = s; bestIdx = v0 + 3; }
            }
            __syncthreads();   // tile[buf] is reused at iteration k+1's issue
        }
    }

    // Tail (empty for V=128000, keeps the kernel shape-general).
    for (int v = ntiles * TILE + t; v < V; v += BLOCK) {
        const float s = fmaf(row[v], invT, gumbel(hash32(bV + (unsigned)v)));
        if (s > best) { best = s; bestIdx = v; }
    }

    // Block-wide argmax reduction in LDS (first-index tie-break like argmax).
    redv[t] = best;
    redi[t] = bestIdx;
    __syncthreads();
#pragma unroll
    for (int s = BLOCK / 2; s > 0; s >>= 1) {
        if (t < s) {
            const float ov = redv[t + s];
            const int   oi = redi[t + s];
            if (ov > redv[t] || (ov == redv[t] && oi < redi[t])) {
                redv[t] = ov;
                redi[t] = oi;
            }
        }
        __syncthreads();
    }
    if (t == 0) out[b] = redi[0];
}

extern "C" void kernel_launch(void* const* d_in, const int* in_sizes, int n_in,
                              void* d_out, int out_size, void* d_ws, size_t ws_size,
                              hipStream_t stream) {
    const float* logits = (const float*)d_in[0];   // [B, V] f32
    const float* temps  = (const float*)d_in[1];   // [B]    f32
    int* out = (int*)d_out;                        // [B] sampled indices
    const int B = in_sizes[1];
    const int V = in_sizes[0] / B;
    Sampler_69922067578951_kernel<<<B, BLOCK, 0, stream>>>(logits, temps, out, V);
}